// Net_SDE_80281528697381
// MI455X (gfx1250) — compile-verified
//
#include <hip/hip_runtime.h>
#include <math.h>

// ---------------------------------------------------------------------------
// Neural-SDE Monte Carlo pricer for MI455X (gfx1250, wave32, WMMA).
// Per wave: one 16-path tile. All MLP GEMMs run on V_WMMA_F32_16X16X4_F32.
// Weights live in LDS transposed ([N][K]) so each B-fragment is ONE
// ds_load_b64; row strides (4 / 68 / 100) chosen so base banks step by 4,
// keeping both lane halves (+0 / +2 banks) conflict-free on the 64 banks.
// Scalar MLP outputs are redistributed through a 16-float per-wave LDS
// scratch (DS is in-order per wave) instead of ds_bpermute chains.
// ---------------------------------------------------------------------------

typedef float v2f __attribute__((ext_vector_type(2)));
typedef float v8f __attribute__((ext_vector_type(8)));

#define MC_SAMP 16384
#define NSTEPS  360
#define WAVES   4
#define BLOCKT  (WAVES * 32)          // 128 threads = 4 wave32 = 1 wave / SIMD32
#define NTILES  (MC_SAMP / 16)        // 1024 path tiles
#define NBLK    (NTILES / WAVES)      // 256 blocks
#define ASTR    101                   // activation row stride (odd -> conflict-free b32)

// ------------------------- LDS layout (float offsets) ----------------------
// All weight matrices stored TRANSPOSED: Wt[n][k].
constexpr int CV_W1 = 0;                 // [100][4]   (K padded 1->4, zero)
constexpr int CV_W2 = CV_W1 + 400;       // [100][100]
constexpr int CV_W3 = CV_W2 + 10000;
constexpr int CV_W4 = CV_W3 + 10000;
constexpr int CV_W5 = CV_W4 + 10000;     // [100]   (100x1 output column)
constexpr int CV_B  = CV_W5 + 104;       // 4 x [100] biases (layers 1..4)
constexpr int CV_B5 = CV_B + 400;        // [4]
constexpr int NET0  = CV_B5 + 4;
constexpr int NET_W1 = 0;                // [64][4]    (K padded to 4, zero)
constexpr int NET_W2 = 256;              // [64][68]   (row stride 68, cols 64..67 zero)
constexpr int NET_W3 = 4608;             // [64]    (64x1 output column)
constexpr int NET_B1 = 4672;             // [64]
constexpr int NET_B2 = 4736;             // [64]
constexpr int NET_B3 = 4800;             // [4]
constexpr int NET_SZ = 4804;
constexpr int ACT0  = NET0 + 4 * NET_SZ;             // per wave: 2 * 16 * ASTR
constexpr int REDU  = ACT0 + WAVES * 2 * 16 * ASTR;  // 16 floats
constexpr int SCR   = REDU + 16;                     // per wave: 16 floats
constexpr int SMEM_FLOATS = SCR + WAVES * 16;
constexpr size_t SMEM_BYTES = (size_t)SMEM_FLOATS * sizeof(float);   // ~246.6 KB < 320 KB

struct SdeParams {
  const float* S0; const float* V0; const float* rate;
  const float* z;  const float* z1;
  const float* cvW[5]; const float* cvB[5];
  const float* nW1[4]; const float* nW2[4]; const float* nW3[4];
  const float* nB1[4]; const float* nB2[4]; const float* nB3[4];
  float* blockOut;     // NBLK * 4 partial sums
};

// fp32 WMMA: D(16x16) = A(16x4) * B(4x16) + C
__device__ __forceinline__ v8f wmma4(v2f a, v2f b, v8f c) {
  return __builtin_amdgcn_wmma_f32_16x16x4_f32(false, a, false, b, (short)0, c,
                                               false, false);
}

// First layer: A comes straight from registers (K padded to 4).
// Wt stored [N][4] in LDS (zero cols for K padding).
template <int N>
__device__ __forceinline__ void layer_first(const float* __restrict__ Wt,
                                            const float* __restrict__ B, v2f a,
                                            float* __restrict__ actOut, int lane) {
  const int half = lane >> 4;
  const int hk   = half * 2;      // lanes 0-15 hold K=0,1 ; lanes 16-31 hold K=2,3
  const int nl   = lane & 15;
  constexpr int NT = (N + 15) / 16;
#pragma unroll
  for (int nt = 0; nt < NT; ++nt) {
    int n  = nt * 16 + nl;
    int nc = (n < N) ? n : (N - 1);       // clamp pad columns (results masked below)
    v2f b = *(const v2f*)(Wt + nc * 4 + hk);   // one ds_load_b64
    v8f c = {0.f, 0.f, 0.f, 0.f, 0.f, 0.f, 0.f, 0.f};
    c = wmma4(a, b, c);
    float bias = B[nc];
#pragma unroll
    for (int r = 0; r < 8; ++r) {
      float v = fmaxf(c[r] + bias, 0.f);  // ReLU
      if (n < N) actOut[(r + half * 8) * ASTR + n] = v;
    }
  }
}

// Hidden layer: [16xK] @ [KxN] + b, ReLU.  K % 4 == 0.
// Wt is [N][WS] (transposed, row stride WS >= K) in LDS.
template <int K, int N, int WS>
__device__ __forceinline__ void layer_hidden(const float* __restrict__ Wt,
                                             const float* __restrict__ B,
                                             const float* __restrict__ actIn,
                                             float* __restrict__ actOut, int lane) {
  const int half = lane >> 4;
  const int hk   = half * 2;
  const int nl   = lane & 15;
  const int m    = lane & 15;
  // Hoist the A-fragments for this 16-row tile into registers once.
  float a0[K / 4], a1[K / 4];
#pragma unroll
  for (int kc = 0; kc < K / 4; ++kc) {
    a0[kc] = actIn[m * ASTR + kc * 4 + hk];
    a1[kc] = actIn[m * ASTR + kc * 4 + hk + 1];
  }
  constexpr int NT = (N + 15) / 16;
#pragma unroll 2
  for (int nt = 0; nt < NT; ++nt) {
    int n  = nt * 16 + nl;
    int nc = (n < N) ? n : (N - 1);
    const float* wrow = Wt + nc * WS;
    v8f c = {0.f, 0.f, 0.f, 0.f, 0.f, 0.f, 0.f, 0.f};
#pragma unroll
    for (int kc = 0; kc < K / 4; ++kc) {
      v2f a; a.x = a0[kc]; a.y = a1[kc];
      v2f b = *(const v2f*)(wrow + kc * 4 + hk);   // one ds_load_b64
      c = wmma4(a, b, c);
    }
    float bias = B[nc];
#pragma unroll
    for (int r = 0; r < 8; ++r) {
      float v = fmaxf(c[r] + bias, 0.f);
      if (n < N) actOut[(r + half * 8) * ASTR + n] = v;
    }
  }
}

// Output layer: [16xK] @ [Kx1] + b.  Column weights stored as K floats in LDS.
// Column 0 of D lives in lanes 0 (rows 0-7) and 16 (rows 8-15); those lanes
// spill to a 16-float per-wave scratch and every lane reads back its row.
template <int K>
__device__ __forceinline__ float layer_out(const float* __restrict__ Wcol, float bias,
                                           const float* __restrict__ actIn,
                                           float* __restrict__ scr, int lane) {
  const int half = lane >> 4;
  const int hk   = half * 2;
  const int nl   = lane & 15;
  const int m    = lane & 15;
  v8f c = {0.f, 0.f, 0.f, 0.f, 0.f, 0.f, 0.f, 0.f};
#pragma unroll
  for (int kc = 0; kc < K / 4; ++kc) {
    v2f a; a.x = actIn[m * ASTR + kc * 4 + hk];
    a.y = actIn[m * ASTR + kc * 4 + hk + 1];
    v2f bw = *(const v2f*)(Wcol + kc * 4 + hk);   // LDS broadcast (same addr per lane half)
    v2f b; b.x = (nl == 0) ? bw.x : 0.f;
    b.y = (nl == 0) ? bw.y : 0.f;
    c = wmma4(a, b, c);
  }
  if (nl == 0) {
#pragma unroll
    for (int r = 0; r < 8; ++r) scr[half * 8 + r] = c[r];
  }
  __builtin_amdgcn_wave_barrier();   // DS is in-order per wave; just pin the schedule
  return scr[m] + bias;
}

// One small net: [in -> 64 -> 64 -> 1]
__device__ __forceinline__ float eval_small(float* __restrict__ sm, int o, v2f a,
                                            float* actA, float* actB, float* scr,
                                            int lane) {
  layer_first<64>(sm + o + NET_W1, sm + o + NET_B1, a, actA, lane);
  __builtin_amdgcn_wave_barrier();
  layer_hidden<64, 64, 68>(sm + o + NET_W2, sm + o + NET_B2, actA, actB, lane);
  __builtin_amdgcn_wave_barrier();
  float r = layer_out<64>(sm + o + NET_W3, sm[o + NET_B3], actB, scr, lane);
  __builtin_amdgcn_wave_barrier();
  return r;
}

__global__ __launch_bounds__(BLOCKT, 1) void sde_kernel(SdeParams P) {
  extern __shared__ float sm[];
  const int tid  = threadIdx.x;
  const int wv   = tid >> 5;
  const int lane = tid & 31;
  const int m    = lane & 15;
  const int half = lane >> 4;

  // ----- phase 1: zero the whole weight region (covers all padding) -----
  for (int i = tid; i < ACT0; i += BLOCKT) sm[i] = 0.f;
  __syncthreads();

  // ----- phase 2: cooperative (transposing) weight load into LDS -----
  auto cp = [&](int dofs, const float* src, int n) {
    for (int i = tid; i < n; i += BLOCKT) sm[dofs + i] = src[i];
  };
  // cv W0 (1x100) -> [100][4], k = 0 column
  for (int i = tid; i < 100; i += BLOCKT) sm[CV_W1 + i * 4] = P.cvW[0][i];
  // cv W1..W3 (100x100) -> transposed [100][100]
  {
    const int dofs[3] = {CV_W2, CV_W3, CV_W4};
    for (int l = 0; l < 3; ++l) {
      const float* src = P.cvW[1 + l];
      for (int i = tid; i < 10000; i += BLOCKT) {
        int k = i / 100, n = i % 100;
        sm[dofs[l] + n * 100 + k] = src[i];
      }
    }
  }
  cp(CV_W5, P.cvW[4], 100);
  for (int l = 0; l < 4; ++l) cp(CV_B + l * 100, P.cvB[l], 100);
  cp(CV_B5, P.cvB[4], 1);
  const int kinTab[4] = {3, 2, 2, 2};  // diffusion, driftV, diffusionV, diffusionV1
  for (int nn = 0; nn < 4; ++nn) {
    int o = NET0 + nn * NET_SZ;
    int kin = kinTab[nn];
    for (int i = tid; i < kin * 64; i += BLOCKT) {       // (kin x 64) -> [64][4]
      int k = i / 64, n = i % 64;
      sm[o + NET_W1 + n * 4 + k] = P.nW1[nn][i];
    }
    for (int i = tid; i < 4096; i += BLOCKT) {           // (64x64) -> [64][68]
      int k = i >> 6, n = i & 63;
      sm[o + NET_W2 + n * 68 + k] = P.nW2[nn][i];
    }
    cp(o + NET_W3, P.nW3[nn], 64);
    cp(o + NET_B1, P.nB1[nn], 64);
    cp(o + NET_B2, P.nB2[nn], 64);
    cp(o + NET_B3, P.nB3[nn], 1);
  }
  __syncthreads();

  float* actA = sm + ACT0 + wv * (2 * 16 * ASTR);
  float* actB = actA + 16 * ASTR;
  float* scr  = sm + SCR + wv * 16;

  const int tile = blockIdx.x * WAVES + wv;
  const int path = tile * 16 + m;  // lanes 16-31 duplicate lanes 0-15's paths

  float S  = P.S0[0];
  float V  = P.V0[0];
  float cv = 0.f;
  const float rate = P.rate[0];
  const float h    = 1.f / (float)NSTEPS;
  const float sqh  = sqrtf(h);

  for (int step = 0; step < NSTEPS; ++step) {
    const float t   = (float)step * h;
    const float dW  = P.z[path * NSTEPS + step] * sqh;
    const float dW1 = P.z1[path * NSTEPS + step] * sqh;
    if (step + 1 < NSTEPS) {  // global_prefetch_b8 for next step's Brownian increments
      __builtin_prefetch(&P.z[path * NSTEPS + step + 1], 0, 0);
      __builtin_prefetch(&P.z1[path * NSTEPS + step + 1], 0, 0);
    }

    // ---- control variate net on [S] (uses current S) ----
    v2f acv; acv.x = half ? 0.f : S; acv.y = 0.f;
    layer_first<100>(sm + CV_W1, sm + CV_B, acv, actA, lane);
    __builtin_amdgcn_wave_barrier();
    layer_hidden<100, 100, 100>(sm + CV_W2, sm + CV_B + 100, actA, actB, lane);
    __builtin_amdgcn_wave_barrier();
    layer_hidden<100, 100, 100>(sm + CV_W3, sm + CV_B + 200, actB, actA, lane);
    __builtin_amdgcn_wave_barrier();
    layer_hidden<100, 100, 100>(sm + CV_W4, sm + CV_B + 300, actA, actB, lane);
    __builtin_amdgcn_wave_barrier();
    const float cvo = layer_out<100>(sm + CV_W5, sm[CV_B5], actB, scr, lane);
    __builtin_amdgcn_wave_barrier();
    cv += cvo * dW;

    // ---- diffusion net on [t, S, V] ----
    v2f ad; ad.x = half ? V : t; ad.y = half ? 0.f : S;
    const float dif = eval_small(sm, NET0 + 0 * NET_SZ, ad, actA, actB, scr, lane);

    // ---- three vol nets on [t, V] ----
    v2f av; av.x = half ? 0.f : t; av.y = half ? 0.f : V;
    const float g1 = eval_small(sm, NET0 + 1 * NET_SZ, av, actA, actB, scr, lane);  // driftV
    const float g2 = eval_small(sm, NET0 + 2 * NET_SZ, av, actA, actB, scr, lane);  // diffusionV
    const float g3 = eval_small(sm, NET0 + 3 * NET_SZ, av, actA, actB, scr, lane);  // diffusionV1

    const float Sn = S + S * rate * h + dif * dW;
    const float Vn = V + g1 * h + g2 * dW + g3 * dW1;
    S = fmaxf(Sn, 0.f);
    V = Vn;
  }

  // ---- deterministic reduction: per-block partial sums ----
  const float payoff = fmaxf(S - 100.f, 0.f);
  const float price  = payoff - cv;
  float r0 = (lane < 16) ? price : 0.f;
  float r1 = (lane < 16) ? price * price : 0.f;
  float r2 = (lane < 16) ? payoff : 0.f;
  float r3 = (lane < 16) ? payoff * payoff : 0.f;
  for (int off = 16; off > 0; off >>= 1) {
    r0 += __shfl_down(r0, off, 32);
    r1 += __shfl_down(r1, off, 32);
    r2 += __shfl_down(r2, off, 32);
    r3 += __shfl_down(r3, off, 32);
  }
  if (lane == 0) {
    float* red = sm + REDU + wv * 4;
    red[0] = r0; red[1] = r1; red[2] = r2; red[3] = r3;
  }
  __syncthreads();
  if (tid == 0) {
    float s0 = 0.f, s1 = 0.f, s2 = 0.f, s3 = 0.f;
    for (int w = 0; w < WAVES; ++w) {
      s0 += sm[REDU + w * 4 + 0];
      s1 += sm[REDU + w * 4 + 1];
      s2 += sm[REDU + w * 4 + 2];
      s3 += sm[REDU + w * 4 + 3];
    }
    float* o = P.blockOut + blockIdx.x * 4;
    o[0] = s0; o[1] = s1; o[2] = s2; o[3] = s3;
  }
}

__global__ void finalize_kernel(const float* __restrict__ blockOut, float* __restrict__ out) {
  if (threadIdx.x == 0 && blockIdx.x == 0) {
    double s = 0.0, s2 = 0.0, pa = 0.0, pa2 = 0.0;
    for (int i = 0; i < NBLK; ++i) {
      s   += (double)blockOut[i * 4 + 0];
      s2  += (double)blockOut[i * 4 + 1];
      pa  += (double)blockOut[i * 4 + 2];
      pa2 += (double)blockOut[i * 4 + 3];
    }
    const double n = (double)MC_SAMP;
    out[0] = (float)(s / n);                               // mean(price)
    out[1] = (float)((s2 - s * s / n) / (n - 1.0));        // var(price, ddof=1)
    out[2] = (float)((pa2 - pa * pa / n) / (n - 1.0));     // var(payoff, ddof=1)
  }
}

// ---------------------------------------------------------------------------
// d_in order (setup_inputs dict order; params pytree flattened with sorted
// dict keys, 'W' list before 'b' list):
//  0:S0 1:V0 2:rate 3:BS_vol 4:indices 5:z 6:z1 7:MC_samples
//  8-12:  cv W0..W4      13-17: cv b0..b4
// 18-20: diffusion W0..W2   21-23: diffusion b0..b2
// 24-26: diffusionV W..     27-29: diffusionV b..
// 30-32: diffusionV1 W..    33-35: diffusionV1 b..
// 36-38: driftV W..         39-41: driftV b..
// ---------------------------------------------------------------------------
extern "C" void kernel_launch(void* const* d_in, const int* in_sizes, int n_in,
                              void* d_out, int out_size, void* d_ws, size_t ws_size,
                              hipStream_t stream) {
  (void)in_sizes; (void)n_in; (void)out_size; (void)ws_size;
  SdeParams P;
  P.S0   = (const float*)d_in[0];
  P.V0   = (const float*)d_in[1];
  P.rate = (const float*)d_in[2];
  P.z    = (const float*)d_in[5];
  P.z1   = (const float*)d_in[6];
  for (int i = 0; i < 5; ++i) {
    P.cvW[i] = (const float*)d_in[8 + i];
    P.cvB[i] = (const float*)d_in[13 + i];
  }
  // kernel net order: 0=diffusion, 1=driftV, 2=diffusionV, 3=diffusionV1
  const int base[4] = {18, 36, 24, 30};
  for (int i = 0; i < 4; ++i) {
    P.nW1[i] = (const float*)d_in[base[i] + 0];
    P.nW2[i] = (const float*)d_in[base[i] + 1];
    P.nW3[i] = (const float*)d_in[base[i] + 2];
    P.nB1[i] = (const float*)d_in[base[i] + 3];
    P.nB2[i] = (const float*)d_in[base[i] + 4];
    P.nB3[i] = (const float*)d_in[base[i] + 5];
  }
  P.blockOut = (float*)d_ws;

  sde_kernel<<<dim3(NBLK), dim3(BLOCKT), SMEM_BYTES, stream>>>(P);
  finalize_kernel<<<dim3(1), dim3(64), 0, stream>>>((const float*)d_ws, (float*)d_out);
}